// DANdiffusionModel_3358664426344
// MI455X (gfx1250) — compile-verified
//
#include <hip/hip_runtime.h>
#include <hip/hip_bf16.h>

#define BATCH 256
#define SEQ 64
#define NH 128
#define NWE 64
#define NB_NODES 10000
#define NCOL (NB_NODES + 1)
#define NB_STEPS 40
#define T_CONSTF 200.0f
#define NEG_BIG -1000000000.0f
#define NCHUNK ((NCOL + 63) / 64)   /* 157 */

typedef float v2f __attribute__((ext_vector_type(2)));
typedef float v8f __attribute__((ext_vector_type(8)));

// ---------------------------------------------------------------------------
// Generic WMMA GEMM:  C[256 x 128] = act( A[256 x K] @ W[K x 128] + bias )
// A rows optionally gathered through idx (embedding lookup).
// act: 0 = none, 1 = elu, 2 = sigmoid
// One wave computes one 16x16 f32 tile via V_WMMA_F32_16X16X4_F32 (K-loop of 4).
// ---------------------------------------------------------------------------
__global__ void gemm_nh_kernel(const float* __restrict__ A, int lda,
                               const int* __restrict__ idx, int idx_stride,
                               const float* __restrict__ W,
                               const float* __restrict__ bias,
                               float* __restrict__ C, int ldc,
                               int K, int act)
{
    const int lane = threadIdx.x & 31;
    const int wave = threadIdx.x >> 5;
    const int half = lane >> 4;      // 0: lanes 0-15, 1: lanes 16-31
    const int l15  = lane & 15;
    const int m0 = blockIdx.x * 16;
    const int n0 = blockIdx.y * 64 + wave * 16;

    int arow = m0 + l15;
    if (idx) arow = idx[(long)arow * idx_stride];
    const float* __restrict__ Arow = A + (long)arow * lda;
    const int col = n0 + l15;

    v8f acc = {};
    for (int k = 0; k < K; k += 4) {
        v2f a, b;
        a.x = Arow[k + 2 * half + 0];              // A: 16x4 slice, K split by half-wave
        a.y = Arow[k + 2 * half + 1];
        b.x = W[(long)(k + 2 * half + 0) * NH + col];
        b.y = W[(long)(k + 2 * half + 1) * NH + col];
        acc = __builtin_amdgcn_wmma_f32_16x16x4_f32(false, a, false, b,
                                                    (short)0, acc, false, false);
    }
    const float bv = bias ? bias[col] : 0.0f;
#pragma unroll
    for (int r = 0; r < 8; ++r) {
        int row = m0 + r + half * 8;
        float v = acc[r] + bv;
        if (act == 1)      v = v > 0.0f ? v : (expf(v) - 1.0f);   // elu
        else if (act == 2) v = 1.0f / (1.0f + expf(-v));          // sigmoid
        C[(long)row * ldc + col] = v;
    }
}

// ---------------------------------------------------------------------------
// Big GEMM tile + fused partial log-softmax statistics:
//   ninf[16 x 64] tile of X2[256,128] @ W_ni2[128,10001] + b_ni2 + discard
//   -> per-row (max, sum exp(v-max)) partials per 64-col chunk.
// Column addresses are clamped (not masked) so EXEC stays all-ones for WMMA.
// ---------------------------------------------------------------------------
__global__ void ninf_tile_kernel(const float* __restrict__ X,
                                 const float* __restrict__ W,
                                 const float* __restrict__ bias,
                                 const float* __restrict__ discard,
                                 float* __restrict__ partials)
{
    __shared__ float tile[16 * 64];
    const int lane = threadIdx.x & 31;
    const int wave = threadIdx.x >> 5;
    const int half = lane >> 4;
    const int l15  = lane & 15;
    const int m0 = blockIdx.x * 16;
    const int col = blockIdx.y * 64 + wave * 16 + l15;
    const int ccol = col < NCOL ? col : NCOL - 1;   // clamp, keep EXEC uniform

    const float* __restrict__ Xrow = X + (long)(m0 + l15) * NH;
    v8f acc = {};
    for (int k = 0; k < NH; k += 4) {
        v2f a, b;
        a.x = Xrow[k + 2 * half + 0];
        a.y = Xrow[k + 2 * half + 1];
        b.x = W[(long)(k + 2 * half + 0) * NCOL + ccol];
        b.y = W[(long)(k + 2 * half + 1) * NCOL + ccol];
        acc = __builtin_amdgcn_wmma_f32_16x16x4_f32(false, a, false, b,
                                                    (short)0, acc, false, false);
    }
#pragma unroll
    for (int r = 0; r < 8; ++r) {
        int row = r + half * 8;
        float v = (col < NCOL)
                ? acc[r] + bias[col] + discard[(long)(m0 + row) * NCOL + col]
                : -3.0e38f;
        tile[row * 64 + wave * 16 + l15] = v;
    }
    __syncthreads();
    if (threadIdx.x < 16) {
        int row = threadIdx.x;
        float m = -3.0e38f;
        for (int j = 0; j < 64; ++j) m = fmaxf(m, tile[row * 64 + j]);
        float s = 0.0f;
        for (int j = 0; j < 64; ++j) s += expf(tile[row * 64 + j] - m);
        long p = ((long)(m0 + row) * NCHUNK + blockIdx.y) * 2;
        partials[p + 0] = m;
        partials[p + 1] = s;
    }
}

// ---------------------------------------------------------------------------
// Context vector: c[b,:] = sum_s softmax(Bs[b,:t])_s * lam(b,s) * Us[b,s,:]
// lam(b,s) = W_lam[bucket(times[b,t-1]-times[b,s])] + b_lam
// ---------------------------------------------------------------------------
__global__ void c_kernel(const float* __restrict__ Bs,
                         const float* __restrict__ Us,
                         const float* __restrict__ times,
                         const float* __restrict__ W_lam,
                         const float* __restrict__ b_lam,
                         float* __restrict__ Cout, int t)
{
    __shared__ float wgt[SEQ];
    const int b = blockIdx.x;
    const int tid = threadIdx.x;           // 0..127
    const float* __restrict__ Bsr = Bs + b * SEQ;
    float m = -3.0e38f;
    for (int s = 0; s < t; ++s) m = fmaxf(m, Bsr[s]);
    float denom = 0.0f;
    for (int s = 0; s < t; ++s) denom += expf(Bsr[s] - m);
    const float qprev = times[b * SEQ + (t - 1)];
    if (tid < t) {
        float diff = qprev - times[b * SEQ + tid];
        int bucket = (int)floorf(diff * ((float)NB_STEPS / T_CONSTF));
        bucket = bucket < 0 ? 0 : (bucket > NB_STEPS ? NB_STEPS : bucket);
        float lam = W_lam[bucket] + b_lam[0];
        wgt[tid] = (expf(Bsr[tid] - m) / denom) * lam;
    }
    __syncthreads();
    float acc = 0.0f;
    const float* __restrict__ Ub = Us + (long)b * SEQ * NH + tid;
    for (int s = 0; s < t; ++s) acc += wgt[s] * Ub[(long)s * NH];
    Cout[b * NH + tid] = acc;
}

// ---------------------------------------------------------------------------
// Loss reducer (single block, 256 threads = 256 rows): combine chunk partials
// into logsumexp, gather picked logit, time loss; deterministic LDS reduction.
// ---------------------------------------------------------------------------
__global__ void loss_reduce_kernel(const float* __restrict__ partials,
                                   const float* __restrict__ X2,
                                   const float* __restrict__ Wni2,
                                   const float* __restrict__ bni2,
                                   const float* __restrict__ discard,
                                   const float* __restrict__ Cc,
                                   const float* __restrict__ Wnt,
                                   const float* __restrict__ bnt,
                                   const float* __restrict__ wparam,
                                   const int* __restrict__ testFrom,
                                   const float* __restrict__ times,
                                   const int* __restrict__ infected,
                                   float* __restrict__ out, int t)
{
    __shared__ float r0[BATCH];
    __shared__ float r1[BATCH];
    const int b = threadIdx.x;

    const float* __restrict__ p = partials + (long)b * NCHUNK * 2;
    float M = -3.0e38f;
    for (int j = 0; j < NCHUNK; ++j) M = fmaxf(M, p[2 * j]);
    float S = 0.0f;
    for (int j = 0; j < NCHUNK; ++j) S += p[2 * j + 1] * expf(p[2 * j] - M);
    float lse = M + logf(S);

    float quand = (t < SEQ) ? times[b * SEQ + t] : -1.0f;
    bool active = quand >= 0.0f;
    int qui = (t < SEQ && active) ? infected[b * SEQ + t] : NB_NODES;

    float picked = bni2[qui] + discard[(long)b * NCOL + qui];
    const float* __restrict__ x2r = X2 + b * NH;
    for (int k = 0; k < NH; ++k) picked += x2r[k] * Wni2[(long)k * NCOL + qui];
    float cr = lse - picked;   // = -lsm[qui]

    float tprev = times[b * SEQ + (t - 1)];
    float mask_prev = (tprev >= 0.0f) ? 1.0f : 0.0f;
    float tf = (float)testFrom[0];
    float mask = ((quand >= tf) || (quand < 0.0f)) ? mask_prev : 0.0f;

    float tt = active ? quand : T_CONSTF;
    float gapT = tt - tprev;
    float gp = logf(gapT + 1.0f);
    float ntdot = bnt[0];
    const float* __restrict__ crow = Cc + b * NH;
    for (int k = 0; k < NH; ++k) ntdot += crow[k] * Wnt[k];
    float nt = 1.0f / (1.0f + expf(-ntdot));
    float w = wparam[0];
    float ntt = nt + w * gp;
    float nt2 = ntt + expf(nt) / w - expf(ntt) / w;
    float tmask = mask * ((quand > 0.0f) ? 1.0f : 0.0f);

    r0[b] = cr * mask;
    r1[b] = -(nt2 * tmask);
    __syncthreads();
    if (b == 0) {
        float s0 = 0.0f, s1 = 0.0f;
        for (int i = 0; i < BATCH; ++i) { s0 += r0[i]; s1 += r1[i]; }
        out[0] += s0;
        out[1] += s1;
    }
}

// ---------------------------------------------------------------------------
// Attention + gated update: u = g*h + (1-g)*d (reads cached H2s = Hs@W_h2+b).
// One block per batch row, 128 threads = NH dims.
// ---------------------------------------------------------------------------
__global__ void attn_kernel(const float* __restrict__ Hs,
                            const float* __restrict__ H2s,
                            const float* __restrict__ H1,
                            const float* __restrict__ Wg,
                            const float* __restrict__ bg,
                            float* __restrict__ Us, int t)
{
    const int b = blockIdx.x;
    const int j = threadIdx.x;
    __shared__ float h1s[NH];
    __shared__ float scores[SEQ];
    __shared__ float red[NH];
    const long base = (long)b * SEQ * NH;
    float hj = Hs[base + (long)t * NH + j];
    if (t == 0) { Us[base + j] = hj; return; }

    h1s[j] = H1[b * NH + j];
    __syncthreads();
    for (int s = j; s < t; s += NH) {
        float d = 0.0f;
        const float* __restrict__ h2 = H2s + base + (long)s * NH;
        for (int k = 0; k < NH; ++k) d += h1s[k] * h2[k];
        scores[s] = d;
    }
    __syncthreads();
    float m = -3.0e38f;
    for (int s = 0; s < t; ++s) m = fmaxf(m, scores[s]);
    float denom = 0.0f;
    for (int s = 0; s < t; ++s) denom += expf(scores[s] - m);
    float dj = 0.0f;
    for (int s = 0; s < t; ++s)
        dj += (expf(scores[s] - m) / denom) * Hs[base + (long)s * NH + j];

    red[j] = hj * Wg[j] + dj * Wg[NH + j];
    __syncthreads();
    for (int off = NH / 2; off > 0; off >>= 1) {
        if (j < off) red[j] += red[j + off];
        __syncthreads();
    }
    float g = 1.0f / (1.0f + expf(-(red[0] + bg[0])));
    Us[base + (long)t * NH + j] = g * hj + (1.0f - g) * dj;
}

// bb[b] = dot(bu[b], W_b) + b_b  -> Bs[b, t]
__global__ void bb_kernel(const float* __restrict__ BU,
                          const float* __restrict__ Wb,
                          const float* __restrict__ bbp,
                          float* __restrict__ Bs, int t)
{
    const int b = blockIdx.x;
    const int j = threadIdx.x;
    __shared__ float red[NH];
    red[j] = BU[b * NH + j] * Wb[j];
    __syncthreads();
    for (int off = NH / 2; off > 0; off >>= 1) {
        if (j < off) red[j] += red[j + off];
        __syncthreads();
    }
    if (j == 0) Bs[b * SEQ + t] = red[0] + bbp[0];
}

__global__ void discard_kernel(const float* __restrict__ times,
                               const int* __restrict__ infected,
                               float* __restrict__ discard, int t)
{
    int b = threadIdx.x;
    if (b < BATCH && times[b * SEQ + t] >= 0.0f)
        discard[(long)b * NCOL + infected[b * SEQ + t]] = NEG_BIG;
}

__global__ void init_kernel(float* __restrict__ discard, float* __restrict__ out)
{
    long i = (long)blockIdx.x * blockDim.x + threadIdx.x;
    if (i < (long)BATCH * NCOL) discard[i] = 0.0f;
    if (i < 2) out[i] = 0.0f;
}

// ---------------------------------------------------------------------------
extern "C" void kernel_launch(void* const* d_in, const int* in_sizes, int n_in,
                              void* d_out, int out_size, void* d_ws, size_t ws_size,
                              hipStream_t stream)
{
    (void)in_sizes; (void)n_in; (void)out_size; (void)ws_size;
    const int*   infected = (const int*)  d_in[0];
    const float* times    = (const float*)d_in[1];
    const float* E        = (const float*)d_in[2];
    const float* W_enc    = (const float*)d_in[3];
    const float* b_enc    = (const float*)d_in[4];
    const float* W_h1     = (const float*)d_in[5];
    const float* b_h1     = (const float*)d_in[6];
    const float* W_h2     = (const float*)d_in[7];
    const float* b_h2     = (const float*)d_in[8];
    const float* W_g      = (const float*)d_in[9];
    const float* b_g      = (const float*)d_in[10];
    const float* W_u      = (const float*)d_in[11];
    const float* b_u      = (const float*)d_in[12];
    const float* W_b      = (const float*)d_in[13];
    const float* b_b      = (const float*)d_in[14];
    const float* W_ni0    = (const float*)d_in[15];
    const float* b_ni0    = (const float*)d_in[16];
    const float* W_ni1    = (const float*)d_in[17];
    const float* b_ni1    = (const float*)d_in[18];
    const float* W_ni2    = (const float*)d_in[19];
    const float* b_ni2    = (const float*)d_in[20];
    const float* W_nt     = (const float*)d_in[21];
    const float* b_nt     = (const float*)d_in[22];
    const float* W_lam    = (const float*)d_in[23];
    const float* b_lam    = (const float*)d_in[24];
    const float* wp       = (const float*)d_in[25];
    const int*   testFrom = (const int*)  d_in[26];
    float* out = (float*)d_out;

    float* ws = (float*)d_ws;
    float* Hs   = ws; ws += (long)BATCH * SEQ * NH;
    float* Us   = ws; ws += (long)BATCH * SEQ * NH;
    float* H2s  = ws; ws += (long)BATCH * SEQ * NH;   // cached Hs @ W_h2 + b_h2
    float* Bs   = ws; ws += BATCH * SEQ;
    float* disc = ws; ws += (long)BATCH * NCOL;
    float* Cbuf = ws; ws += BATCH * NH;
    float* X1   = ws; ws += BATCH * NH;
    float* X2   = ws; ws += BATCH * NH;
    float* H1t  = ws; ws += BATCH * NH;
    float* BUt  = ws; ws += BATCH * NH;
    float* part = ws; ws += (long)BATCH * NCHUNK * 2;

    init_kernel<<<((long)BATCH * NCOL + 255) / 256, 256, 0, stream>>>(disc, out);

    const dim3 gNH(16, 2), blk(128);
    for (int t = 0; t < SEQ; ++t) {
        if (t > 0) {  // loss for step t (uses state from steps < t)
            c_kernel<<<BATCH, NH, 0, stream>>>(Bs, Us, times, W_lam, b_lam, Cbuf, t);
            gemm_nh_kernel<<<gNH, blk, 0, stream>>>(Cbuf, NH, nullptr, 0, W_ni0, b_ni0, X1, NH, NH, 2);
            gemm_nh_kernel<<<gNH, blk, 0, stream>>>(X1, NH, nullptr, 0, W_ni1, b_ni1, X2, NH, NH, 2);
            ninf_tile_kernel<<<dim3(16, NCHUNK), blk, 0, stream>>>(X2, W_ni2, b_ni2, disc, part);
            loss_reduce_kernel<<<1, BATCH, 0, stream>>>(part, X2, W_ni2, b_ni2, disc, Cbuf,
                                                        W_nt, b_nt, wp, testFrom, times, infected, out, t);
        }
        discard_kernel<<<1, BATCH, 0, stream>>>(times, infected, disc, t);
        // h = elu(E[qui] @ W_enc + b_enc) -> Hs[:, t]
        gemm_nh_kernel<<<gNH, blk, 0, stream>>>(E, NWE, infected + t, SEQ, W_enc, b_enc,
                                                Hs + t * NH, SEQ * NH, NWE, 1);
        // h1 = h @ W_h1 + b_h1
        gemm_nh_kernel<<<gNH, blk, 0, stream>>>(Hs + t * NH, SEQ * NH, nullptr, 0, W_h1, b_h1,
                                                H1t, NH, NH, 0);
        attn_kernel<<<BATCH, NH, 0, stream>>>(Hs, H2s, H1t, W_g, b_g, Us, t);
        // bu = elu(u @ W_u + b_u)
        gemm_nh_kernel<<<gNH, blk, 0, stream>>>(Us + t * NH, SEQ * NH, nullptr, 0, W_u, b_u,
                                                BUt, NH, NH, 1);
        bb_kernel<<<BATCH, NH, 0, stream>>>(BUt, W_b, b_b, Bs, t);
        // cache h2_t = h @ W_h2 + b_h2 (avoids reference's O(t) recompute)
        gemm_nh_kernel<<<gNH, blk, 0, stream>>>(Hs + t * NH, SEQ * NH, nullptr, 0, W_h2, b_h2,
                                                H2s + t * NH, SEQ * NH, NH, 0);
    }
    // final loss term (quand = -1, qui = NB_NODES, full history length SEQ)
    {
        const int t = SEQ;
        c_kernel<<<BATCH, NH, 0, stream>>>(Bs, Us, times, W_lam, b_lam, Cbuf, t);
        gemm_nh_kernel<<<gNH, blk, 0, stream>>>(Cbuf, NH, nullptr, 0, W_ni0, b_ni0, X1, NH, NH, 2);
        gemm_nh_kernel<<<gNH, blk, 0, stream>>>(X1, NH, nullptr, 0, W_ni1, b_ni1, X2, NH, NH, 2);
        ninf_tile_kernel<<<dim3(16, NCHUNK), blk, 0, stream>>>(X2, W_ni2, b_ni2, disc, part);
        loss_reduce_kernel<<<1, BATCH, 0, stream>>>(part, X2, W_ni2, b_ni2, disc, Cbuf,
                                                    W_nt, b_nt, wp, testFrom, times, infected, out, t);
    }
}